// GraphSage_90486370992431
// MI455X (gfx1250) — compile-verified
//
#include <hip/hip_runtime.h>

typedef float v2f __attribute__((ext_vector_type(2)));
typedef float v4f __attribute__((ext_vector_type(4)));
typedef float v8f __attribute__((ext_vector_type(8)));

// __has_builtin for target builtins is only meaningful in the device pass;
// the host pass would (correctly) report it missing and abort the build.
#if defined(__HIP_DEVICE_COMPILE__) && defined(__gfx1250__)
#if !__has_builtin(__builtin_amdgcn_wmma_f32_16x16x4_f32)
#error "gfx1250 f32 WMMA builtin __builtin_amdgcn_wmma_f32_16x16x4_f32 not available"
#endif
#endif

#define FEAT 256
#define K2   512          // 2*FEAT
#define LDA  516          // padded LDS row (516 mod 64 = 4 -> conflict-free column reads)

__device__ __forceinline__ v8f wmma_f32(v2f a, v2f b, v8f c) {
  // 8 args: (neg_a, A, neg_b, B, c_mod, C, reuse_a, reuse_b)
  return __builtin_amdgcn_wmma_f32_16x16x4_f32(false, a, false, b, (short)0, c, false, false);
}

__global__ void zero_f32(float* __restrict__ p, int n) {
  int i = blockIdx.x * 256 + threadIdx.x;
  if (i < n) p[i] = 0.0f;
}

// One block = 16 output rows (M) x 256 output cols (N).
// 128 threads = 4 waves; wave w owns N-slab [w*64, w*64+64).
// A row = concat( featRoot[rootIdx[row]],  mean_j featNeigh[neighIdx[row*nNeigh+j]] )
// rootIdx/neighIdx == nullptr => identity (row itself).
// mode 0: out[row*256+col] = relu(v)
// mode 1: out[(row/groupSize)*256+col] += relu(v) * groupScale   (atomic)
__global__ __launch_bounds__(128)
void fused_gather_gemm(const float* __restrict__ featRoot, const int* __restrict__ rootIdx,
                       const float* __restrict__ featNeigh, const int* __restrict__ neighIdx,
                       int nNeigh, float invNeigh,
                       const float* __restrict__ W,        // [256][512] row-major (out,in)
                       float* __restrict__ out,
                       int mode, float groupScale, int groupSize)
{
  __shared__ float A[16][LDA];

  const int  tid      = threadIdx.x;
  const int  rowLocal = tid >> 3;            // 0..15
  const int  seg      = tid & 7;             // 0..7 -> 32-float column segment
  const long mBase    = (long)blockIdx.x * 16;
  const long gRow     = mBase + rowLocal;

  // ---------------- stage A tile (gather + neighbor mean) into LDS ----------------
  {
    const long rnode = rootIdx ? (long)rootIdx[gRow] : gRow;
    const v4f* __restrict__ src = (const v4f*)(featRoot + rnode * FEAT) + seg * 8;
#pragma unroll
    for (int q = 0; q < 8; ++q) {
      v4f v = src[q];
      *(v4f*)&A[rowLocal][seg * 32 + q * 4] = v;
    }

    v4f acc[8];
#pragma unroll
    for (int q = 0; q < 8; ++q) acc[q] = v4f{0.f, 0.f, 0.f, 0.f};
    for (int j = 0; j < nNeigh; ++j) {
      const long nnode = neighIdx ? (long)neighIdx[gRow * nNeigh + j] : gRow;
      const v4f* __restrict__ s2 = (const v4f*)(featNeigh + nnode * FEAT) + seg * 8;
#pragma unroll
      for (int q = 0; q < 8; ++q) acc[q] += s2[q];
    }
#pragma unroll
    for (int q = 0; q < 8; ++q) {
      v4f v = acc[q] * invNeigh;
      *(v4f*)&A[rowLocal][FEAT + seg * 32 + q * 4] = v;
    }
  }
  __syncthreads();

  // ---------------- WMMA GEMM: (16 x 512) x (512 x 256) ----------------
  const int lane = tid & 31;                 // wave32
  const int wave = tid >> 5;                 // 0..3
  const int half = lane >> 4;                // 0: K={k,k+1}, 1: K={k+2,k+3}
  const int lrow = lane & 15;                // M row / N col within a 16-tile
  const int nBase = wave * 64;

  v8f c0 = v8f{0,0,0,0,0,0,0,0};
  v8f c1 = c0, c2 = c0, c3 = c0;

  for (int k0 = 0; k0 < K2; k0 += 4) {
    const int kk = k0 + 2 * half;
    v2f a  = *(const v2f*)&A[lrow][kk];
    v2f b0 = *(const v2f*)&W[(long)(nBase +  0 + lrow) * K2 + kk];
    v2f b1 = *(const v2f*)&W[(long)(nBase + 16 + lrow) * K2 + kk];
    v2f b2 = *(const v2f*)&W[(long)(nBase + 32 + lrow) * K2 + kk];
    v2f b3 = *(const v2f*)&W[(long)(nBase + 48 + lrow) * K2 + kk];
    c0 = wmma_f32(a, b0, c0);
    c1 = wmma_f32(a, b1, c1);
    c2 = wmma_f32(a, b2, c2);
    c3 = wmma_f32(a, b3, c3);
  }

  // ---------------- epilogue: relu + store / grouped atomic mean ----------------
  // C layout: VGPR i, lanes 0-15: (M=i, N=lane); lanes 16-31: (M=8+i, N=lane-16)
  v8f cs[4] = {c0, c1, c2, c3};
#pragma unroll
  for (int nt = 0; nt < 4; ++nt) {
#pragma unroll
    for (int i = 0; i < 8; ++i) {
      float v = cs[nt][i];
      v = v > 0.f ? v : 0.f;
      const int  m   = half * 8 + i;
      const long row = mBase + m;
      const int  col = nBase + nt * 16 + lrow;
      if (mode == 0) {
        out[row * FEAT + col] = v;
      } else {
        __hip_atomic_fetch_add(&out[(row / groupSize) * FEAT + col], v * groupScale,
                               __ATOMIC_RELAXED, __HIP_MEMORY_SCOPE_AGENT);
      }
    }
  }
}

// scores = Ein[4096,256] @ W3[64,256]^T -> out[4096,64]; no relu.
// 256 threads = 8 waves; wave owns one 16-row M tile, full N=64 (4 tiles).
__global__ __launch_bounds__(256)
void gemm_scores(const float* __restrict__ Ein,
                 const float* __restrict__ W3,
                 float* __restrict__ out)
{
  const int tid  = threadIdx.x;
  const int lane = tid & 31;
  const int wave = tid >> 5;
  const int half = lane >> 4;
  const int lrow = lane & 15;
  const long mBase = ((long)blockIdx.x * 8 + wave) * 16;

  v8f c0 = v8f{0,0,0,0,0,0,0,0};
  v8f c1 = c0, c2 = c0, c3 = c0;

  for (int k0 = 0; k0 < 256; k0 += 4) {
    const int kk = k0 + 2 * half;
    v2f a  = *(const v2f*)&Ein[(mBase + lrow) * 256 + kk];
    v2f b0 = *(const v2f*)&W3[(long)( 0 + lrow) * 256 + kk];
    v2f b1 = *(const v2f*)&W3[(long)(16 + lrow) * 256 + kk];
    v2f b2 = *(const v2f*)&W3[(long)(32 + lrow) * 256 + kk];
    v2f b3 = *(const v2f*)&W3[(long)(48 + lrow) * 256 + kk];
    c0 = wmma_f32(a, b0, c0);
    c1 = wmma_f32(a, b1, c1);
    c2 = wmma_f32(a, b2, c2);
    c3 = wmma_f32(a, b3, c3);
  }

  v8f cs[4] = {c0, c1, c2, c3};
#pragma unroll
  for (int nt = 0; nt < 4; ++nt) {
#pragma unroll
    for (int i = 0; i < 8; ++i) {
      const int  m   = half * 8 + i;
      const int  col = nt * 16 + lrow;
      out[(mBase + m) * 64 + col] = cs[nt][i];
    }
  }
}

extern "C" void kernel_launch(void* const* d_in, const int* in_sizes, int n_in,
                              void* d_out, int out_size, void* d_ws, size_t ws_size,
                              hipStream_t stream) {
  const int*   T0  = (const int*)d_in[0];
  const int*   T1  = (const int*)d_in[1];
  const int*   T2  = (const int*)d_in[2];
  const float* emb = (const float*)d_in[3];
  const float* W1  = (const float*)d_in[4];   // [256,512]
  const float* W2  = (const float*)d_in[5];   // [256,512]
  const float* W3  = (const float*)d_in[6];   // [64,256]
  float* scores = (float*)d_out;

  const int B = 4096, F1 = 25, F2 = 10;
  const int M1 = B * F1;                      // 102400

  float* h1buf  = (float*)d_ws;               // [4096,256]
  float* h0buf  = h1buf + (size_t)B * FEAT;   // [4096,256]
  float* embBuf = h0buf + (size_t)B * FEAT;   // [4096,256]

  // h1 accumulator must start at zero every call
  zero_f32<<<(B * FEAT + 255) / 256, 256, 0, stream>>>(h1buf, B * FEAT);

  // h1 path: rows over B*F1; relu then mean over F1 groups (atomic)
  fused_gather_gemm<<<M1 / 16, 128, 0, stream>>>(
      emb, T1, emb, T2, F2, 1.0f / F2, W1, h1buf, 1, 1.0f / F1, F1);

  // h0 path: rows over B; direct relu store
  fused_gather_gemm<<<B / 16, 128, 0, stream>>>(
      emb, T0, emb, T1, F1, 1.0f / F1, W1, h0buf, 0, 1.0f, 1);

  // embedding = relu(concat(h0,h1) @ W2^T)  (identity "gather", nNeigh=1)
  fused_gather_gemm<<<B / 16, 128, 0, stream>>>(
      h0buf, nullptr, h1buf, nullptr, 1, 1.0f, W2, embBuf, 0, 1.0f, 1);

  // scores = embedding @ W3^T
  gemm_scores<<<B / 128, 256, 0, stream>>>(embBuf, W3, scores);
}